// DeltaOrderLoss_18270790877693
// MI455X (gfx1250) — compile-verified
//
#include <hip/hip_runtime.h>

typedef float v2f __attribute__((ext_vector_type(2)));
typedef float v8f __attribute__((ext_vector_type(8)));

#define NROWS 384      // N = 2*bs
#define NM1   383
#define FEAT  1024
#define BS    192
#define DELTA 0.1f

// Row i of the virtual concat [features[:,0]; features[:,1]]
__device__ __forceinline__ const float* feat_row(const float* f, int r) {
    return f + ((r < BS) ? ((size_t)r * (2 * FEAT))
                         : ((size_t)(r - BS) * (2 * FEAT) + FEAT));
}

// ---------------- Kernel 0: per-row squared norms ----------------
__global__ void row_norms_kernel(const float* __restrict__ features,
                                 float* __restrict__ sq) {
    const int i = blockIdx.x;
    const float* row = feat_row(features, i);
    float s = 0.0f;
    for (int d = threadIdx.x; d < FEAT; d += blockDim.x) {
        float v = row[d];
        s += v * v;
    }
    __shared__ float red[256];
    red[threadIdx.x] = s;
    __syncthreads();
    for (int off = 128; off > 0; off >>= 1) {
        if (threadIdx.x < off) red[threadIdx.x] += red[threadIdx.x + off];
        __syncthreads();
    }
    if (threadIdx.x == 0) sq[i] = red[0];
}

// ---------------- Kernel 1: Gram via V_WMMA_F32_16X16X4_F32, fused sqrt-dist epilogue ----
// One wave (32 lanes) per 16x16 output tile. A-frag (16x4 f32): lane L holds
// row M=L&15, K = (L>>4)*2 + {0,1}. B-frag symmetric with N=L&15 (B = feats^T).
__global__ __launch_bounds__(32) void gram_wmma_kernel(
        const float* __restrict__ features,
        const float* __restrict__ sq,
        float* __restrict__ zfull) {
    const int lane = threadIdx.x;            // 0..31
    const int tm   = blockIdx.y * 16;
    const int tn   = blockIdx.x * 16;
    const int sub  = lane & 15;
    const int koff = (lane >> 4) << 1;       // 0 or 2

    const float* rowA = feat_row(features, tm + sub);
    const float* rowB = feat_row(features, tn + sub);

    v8f acc = {};
#pragma unroll 4
    for (int k = 0; k < FEAT; k += 4) {
        v2f a = *(const v2f*)(rowA + k + koff);
        v2f b = *(const v2f*)(rowB + k + koff);
        acc = __builtin_amdgcn_wmma_f32_16x16x4_f32(
            /*neg_a=*/false, a, /*neg_b=*/false, b,
            /*c_mod=*/(short)0, acc, /*reuse_a=*/false, /*reuse_b=*/false);
    }

    // C/D layout: VGPR v -> M = v + (lane>>4)*8, N = lane&15
    const int n     = tn + sub;
    const int mbase = tm + ((lane >> 4) << 3);
    const float sqn = sq[n];
#pragma unroll
    for (int v = 0; v < 8; ++v) {
        const int m  = mbase + v;
        float d2     = sq[m] + sqn - 2.0f * acc[v];
        zfull[m * NROWS + n] = sqrtf(fmaxf(d2, 0.0f));
    }
}

// ---------------- Kernel 2: per-row 383x383 pair loss -------------
__global__ __launch_bounds__(256) void row_loss_kernel(
        const float* __restrict__ zfull,
        const int* __restrict__ labels,
        float* __restrict__ partials) {
    const int i   = blockIdx.x;
    const int tid = threadIdx.x;
    const int bd  = blockDim.x;  // 256

    __shared__ float zrow[NM1];
    __shared__ float rrow[NM1];
    __shared__ int   yrow[NM1];
    __shared__ unsigned long long mbuf[256];
    __shared__ float redp[256];
    __shared__ float redn[256];

    const int li = labels[i % BS];
    unsigned long long mymask = 0ull;
    for (int k = tid; k < NM1; k += bd) {
        const int j = k + (k >= i);          // off-diagonal column mapping
        zrow[k] = zfull[i * NROWS + j];
        int y = labels[j % BS] - li;
        y = (y < 0) ? -y : y;                // |label diff|, exact integer
        yrow[k] = y;
        mymask |= (1ull << y);
    }
    mbuf[tid] = mymask;
    __syncthreads();
    for (int off = 128; off > 0; off >>= 1) {
        if (tid < off) mbuf[tid] |= mbuf[tid + off];
        __syncthreads();
    }
    const unsigned long long mask = mbuf[0];

    // dense rank = #distinct values strictly below
    for (int k = tid; k < NM1; k += bd) {
        const int y = yrow[k];
        rrow[k] = (float)__popcll(mask & ((1ull << y) - 1ull));
    }
    __syncthreads();

    float pos = 0.0f, neg = 0.0f;
    for (int j = 0; j < NM1; ++j) {
        const float zj = zrow[j];
        const float rj = rrow[j];
        const int   yj = yrow[j];
        for (int k = tid; k < NM1; k += bd) {
            const float a = fabsf(zrow[k] - zj);      // abs_diffs[i,j,k]
            if (yrow[k] == yj) {
                const float w = 1.0f / (1.0f + __expf(DELTA - a));
                pos += a * w;
            } else {
                const float margin = fabsf(rrow[k] - rj) * DELTA;
                const float d = a - margin;
                neg += d * d;
            }
        }
    }
    redp[tid] = pos;
    redn[tid] = neg;
    __syncthreads();
    for (int off = 128; off > 0; off >>= 1) {
        if (tid < off) {
            redp[tid] += redp[tid + off];
            redn[tid] += redn[tid + off];
        }
        __syncthreads();
    }
    if (tid == 0) {
        partials[i]         = redp[0];
        partials[NROWS + i] = redn[0];
    }
}

// ---------------- Kernel 3: deterministic final reduce ------------
__global__ void final_reduce_kernel(const float* __restrict__ partials,
                                    float* __restrict__ out) {
    __shared__ float red[256];
    float s = 0.0f;
    for (int i = threadIdx.x; i < 2 * NROWS; i += blockDim.x) s += partials[i];
    red[threadIdx.x] = s;
    __syncthreads();
    for (int off = 128; off > 0; off >>= 1) {
        if (threadIdx.x < off) red[threadIdx.x] += red[threadIdx.x + off];
        __syncthreads();
    }
    if (threadIdx.x == 0)
        out[0] = red[0] * (1.0f / ((float)NROWS * (float)NM1 * (float)NM1));
}

extern "C" void kernel_launch(void* const* d_in, const int* in_sizes, int n_in,
                              void* d_out, int out_size, void* d_ws, size_t ws_size,
                              hipStream_t stream) {
    const float* features = (const float*)d_in[0];   // [192, 2, 1024] f32
    const int*   labels   = (const int*)d_in[1];     // [192, 1] i32
    // d_in[2] (ranks) is a dead argument.
    float* out = (float*)d_out;

    // Workspace layout (floats): sq[384] | zfull[384*384] | partials[768]
    float* sq       = (float*)d_ws;
    float* zfull    = sq + NROWS;
    float* partials = zfull + NROWS * NROWS;

    row_norms_kernel<<<NROWS, 256, 0, stream>>>(features, sq);
    gram_wmma_kernel<<<dim3(NROWS / 16, NROWS / 16), 32, 0, stream>>>(features, sq, zfull);
    row_loss_kernel<<<NROWS, 256, 0, stream>>>(zfull, labels, partials);
    final_reduce_kernel<<<1, 256, 0, stream>>>(partials, out);
}